// OctreeFeatureExtractor_56453050138965
// MI455X (gfx1250) — compile-verified
//
#include <hip/hip_runtime.h>
#include <cstdint>

#define TILE 256
#define IMG_W 256

// ---------------------------------------------------------------------------
// Pre-pass: interleave depth+mask so each corner gather is ONE b64 load.
// Maps are 8*256*256 = 524288 px -> 4 MB packed, lives entirely in L2.
// ---------------------------------------------------------------------------
__global__ __launch_bounds__(256) void pack_maps_kernel(
    const float* __restrict__ mask, const float* __restrict__ depth,
    float2* __restrict__ packed, int n) {
  int i = blockIdx.x * 256 + threadIdx.x;
  if (i < n) packed[i] = make_float2(depth[i], mask[i]);  // x=depth, y=mask
}

// ---------------------------------------------------------------------------
// Main kernel. One thread = one voxel. pts/batch_id tiles are staged into
// LDS with the gfx1250 async copy engine (GLOBAL_LOAD_ASYNC_TO_LDS_B128),
// tracked by ASYNCcnt.
// ---------------------------------------------------------------------------
template <bool USE_PACKED>
__global__ __launch_bounds__(256) void voxel_cover_free_kernel(
    const float* __restrict__ pts, const int* __restrict__ batch_id,
    const float* __restrict__ Kmat, const float* __restrict__ grid_size,
    const float2* __restrict__ packed, const float* __restrict__ mask,
    const float* __restrict__ depth, float2* __restrict__ out, int N) {
  __shared__ float sh_pts[TILE * 3];  // 3072 B = 192 chunks of 16 B
  __shared__ int   sh_bid[TILE];      // 1024 B =  64 chunks of 16 B

  const int t = threadIdx.x;
  const int tile = blockIdx.x * TILE;
  const int tileVox = min(TILE, N - tile);

  // ---- async stage-in (CDNA5: global_load_async_to_lds, ASYNCcnt) ----
  {
    const int ptsBytes = tileVox * 12;
    const int bidBytes = tileVox * 4;
    if (t < 192) {
      const int off = t * 16;
      if (off < ptsBytes) {
        const unsigned lds  = (unsigned)(uintptr_t)((const char*)sh_pts + off);
        const unsigned voff = (unsigned)(tile * 12 + off);  // <= 12e6, fits
        asm volatile("global_load_async_to_lds_b128 %0, %1, %2"
                     :: "v"(lds), "v"(voff), "s"(pts) : "memory");
      }
    } else {
      const int off = (t - 192) * 16;
      if (off < bidBytes) {
        const unsigned lds  = (unsigned)(uintptr_t)((const char*)sh_bid + off);
        const unsigned voff = (unsigned)(tile * 4 + off);
        asm volatile("global_load_async_to_lds_b128 %0, %1, %2"
                     :: "v"(lds), "v"(voff), "s"(batch_id) : "memory");
      }
    }
#if __has_builtin(__builtin_amdgcn_s_wait_asynccnt)
    __builtin_amdgcn_s_wait_asynccnt(0);
#else
    asm volatile("s_wait_asynccnt 0x0" ::: "memory");
#endif
  }
  __syncthreads();  // each wave waited its own copies; barrier makes all visible

  if (t >= tileVox) return;

  const float x0 = sh_pts[t * 3 + 0];
  const float y0 = sh_pts[t * 3 + 1];
  const float z0 = sh_pts[t * 3 + 2];
  const int   b  = sh_bid[t];

  // uniform scalars (uniform addresses -> scalar loads)
  const float gs = grid_size[0];
  const float fx = Kmat[0], cx = Kmat[2];
  const float fy = Kmat[4], cy = Kmat[5];
  const float h = 0.5f * gs;  // exact (power-of-two scale), matches gs*(-0.5)

  // only 2 distinct values per axis across the 8 corners
  const float xm = x0 - h, xp = x0 + h;
  const float ym = y0 - h, yp = y0 + h;
  const float zm = fmaxf(z0 - h, 1e-6f);  // reference clamps z AFTER offset
  const float zp = fmaxf(z0 + h, 1e-6f);

  // Bit-exact pixel selection vs reference: IEEE f32 div, rintf = round-half-
  // even (jnp.round), clamp then convert.
  auto pix = [](float f, float c, float a, float z) -> int {
    float u = f * a / z + c;                       // exact IEEE mul/div/add
    float r = fminf(fmaxf(rintf(u), 0.0f), 255.0f);
    return (int)r;
  };
  const int u_mm = pix(fx, cx, xm, zm);  // (x-, z-)
  const int u_pm = pix(fx, cx, xp, zm);  // (x+, z-)
  const int u_mp = pix(fx, cx, xm, zp);  // (x-, z+)
  const int u_pp = pix(fx, cx, xp, zp);  // (x+, z+)
  const int v_mm = pix(fy, cy, ym, zm);
  const int v_pm = pix(fy, cy, yp, zm);
  const int v_mp = pix(fy, cy, ym, zp);
  const int v_pp = pix(fy, cy, yp, zp);

  const int base = b << 16;  // b * 256 * 256
  float cover = 0.0f, freeacc = 0.0f;

  // FACE_OFFSET vertex multiplicities: corner k weight = (k even ? 5 : 4)
  auto corner = [&](int pu, int pv, float zc, float w) {
    const int idx = base + (pv << 8) + pu;
    float d, m;
    if (USE_PACKED) { float2 dm = packed[idx]; d = dm.x; m = dm.y; }
    else            { d = depth[idx]; m = mask[idx]; }
    // sigmoid((d-z)/tau) = 1/(1+exp((z-d)/tau)); 1/0.05f rounds to exactly 20
    const float sg = 1.0f / (1.0f + __expf((zc - d) * 20.0f));
    cover   += w * m;
    freeacc += w * sg;
  };
  corner(u_mm, v_mm, zm, 5.0f);  // (-,-,-)
  corner(u_pm, v_mm, zm, 4.0f);  // (+,-,-)
  corner(u_pm, v_pm, zm, 5.0f);  // (+,+,-)
  corner(u_mm, v_pm, zm, 4.0f);  // (-,+,-)
  corner(u_mp, v_mp, zp, 5.0f);  // (-,-,+)
  corner(u_pp, v_mp, zp, 4.0f);  // (+,-,+)
  corner(u_pp, v_pp, zp, 5.0f);  // (+,+,+)
  corner(u_mp, v_pp, zp, 4.0f);  // (-,+,+)

  const float c36 = 1.0f / 36.0f;
  out[tile + t] = make_float2(cover * c36, freeacc * c36);  // b64 store
}

// ---------------------------------------------------------------------------
extern "C" void kernel_launch(void* const* d_in, const int* in_sizes, int n_in,
                              void* d_out, int out_size, void* d_ws, size_t ws_size,
                              hipStream_t stream) {
  const float* pts   = (const float*)d_in[0];
  const float* mask  = (const float*)d_in[1];
  const float* depth = (const float*)d_in[2];
  const float* Kmat  = (const float*)d_in[3];
  const int*   bid   = (const int*)d_in[4];
  const float* gs    = (const float*)d_in[5];

  const int N    = in_sizes[0] / 3;   // NUM_VOXELS
  const int npix = in_sizes[1];       // BATCH*H*W
  float2* out2 = (float2*)d_out;
  const int blocks = (N + TILE - 1) / TILE;

  if (ws_size >= (size_t)npix * sizeof(float2)) {
    float2* packed = (float2*)d_ws;
    pack_maps_kernel<<<(npix + 255) / 256, 256, 0, stream>>>(mask, depth, packed, npix);
    voxel_cover_free_kernel<true><<<blocks, TILE, 0, stream>>>(
        pts, bid, Kmat, gs, packed, mask, depth, out2, N);
  } else {
    voxel_cover_free_kernel<false><<<blocks, TILE, 0, stream>>>(
        pts, bid, Kmat, gs, nullptr, mask, depth, out2, N);
  }
}